// CIGINModel_1529008357940
// MI455X (gfx1250) — compile-verified
//
#include <hip/hip_runtime.h>
#include <hip/hip_bf16.h>

#define NN   8192     // nodes per side
#define DD   42       // hidden dim
#define DP   48       // padded hidden dim
#define EE   65536    // edges per side
#define EIND 10       // edge feature dim
#define BB   32       // graphs
#define NPGD 256      // nodes per graph
#define D2D  84       // 2*D
#define WED  1764     // D*D
#define WEP  1792     // padded D*D (112 * 16)
#define KP2  44       // padded K for the We GEMM (42 -> 44)

typedef __attribute__((ext_vector_type(2))) float v2f;
typedef __attribute__((ext_vector_type(8))) float v8f;

__device__ __forceinline__ v8f wmma4(v2f a, v2f b, v8f c) {
  // D = A(16x4, f32) x B(4x16, f32) + C(16x16, f32)
  return __builtin_amdgcn_wmma_f32_16x16x4_f32(false, a, false, b, (short)0, c,
                                               false, false);
}

// ---------------------------------------------------------------- utilities
__global__ void fillz(float* p, size_t n) {
  size_t i = (size_t)blockIdx.x * blockDim.x + threadIdx.x;
  size_t st = (size_t)gridDim.x * blockDim.x;
  for (; i < n; i += st) p[i] = 0.f;
}

// Y[m,n] = act(sum_k X[m*ldx+k]*W[n*K+k] + b[n]);  Y stride ldy
__global__ void dense(const float* __restrict__ X, int ldx,
                      const float* __restrict__ W, const float* __restrict__ b,
                      float* __restrict__ Y, int ldy, int M, int N, int K,
                      int relu_flag) {
  int idx = blockIdx.x * blockDim.x + threadIdx.x;
  if (idx >= M * N) return;
  int m = idx / N, n = idx % N;
  const float* x = X + (size_t)m * ldx;
  const float* w = W + (size_t)n * K;
  float acc = b[n];
  for (int k = 0; k < K; ++k) acc += x[k] * w[k];
  if (relu_flag) acc = fmaxf(acc, 0.f);
  Y[(size_t)m * ldy + n] = acc;
}

// zero-padded copy of en2_W: W2p[f*44 + k] = W2[f*42 + k] (f<1764, k<42)
__global__ void pad_w2(const float* __restrict__ W2, float* __restrict__ W2p) {
  int idx = blockIdx.x * blockDim.x + threadIdx.x;
  if (idx >= WEP * KP2) return;
  int f = idx / KP2, k = idx % KP2;
  W2p[idx] = (f < WED && k < DD) ? W2[(size_t)f * DD + k] : 0.f;
}

// ---------------------------------------------------- We = bf16(h1 @ en2^T)
// stored transposed within each edge row: We[e*WEP + o*42 + i] = We(e,i,o)
__global__ __launch_bounds__(256) void we_gemm(
    const float* __restrict__ h1 /*E x DP, pad 0*/,
    const float* __restrict__ W2p /*WEP x KP2, pad 0*/,
    const float* __restrict__ b2, __hip_bfloat16* __restrict__ We) {
  int wave = threadIdx.x >> 5, lane = threadIdx.x & 31;
  int half = lane >> 4, l15 = lane & 15;
  int e0 = blockIdx.x * 16;
  int f0 = (blockIdx.y * 8 + wave) * 16;
  int f = f0 + l15;
  v8f acc = {};
  const size_t arow = (size_t)(e0 + l15) * DP;
  const size_t brow = (size_t)f * KP2;
#pragma unroll
  for (int k0 = 0; k0 < KP2; k0 += 4) {
    int kb = k0 + half * 2;
    v2f a, b;
    a[0] = h1[arow + kb];
    a[1] = h1[arow + kb + 1];
    b[0] = W2p[brow + kb];
    b[1] = W2p[brow + kb + 1];
    acc = wmma4(a, b, acc);
  }
  if (f < WED) {
    int o = f % DD, i = f / DD;
    size_t tpos = (size_t)o * DD + i;  // transposed position
    float bias = b2[f];
#pragma unroll
    for (int r = 0; r < 8; ++r) {
      int e = e0 + r + half * 8;
      We[(size_t)e * WEP + tpos] = __float2bfloat16(acc[r] + bias);
    }
  }
}

// ------------------------------------------- per-edge message + segment sum
__global__ __launch_bounds__(256) void edge_msg(
    const float* __restrict__ cur /*N x DP*/,
    const __hip_bfloat16* __restrict__ We, const int* __restrict__ src,
    const int* __restrict__ dst, float* __restrict__ msg /*N x DP*/) {
  __shared__ float xs[4][DD];
  int grp = threadIdx.x >> 6;  // 4 edges per block, 64 threads each
  int t = threadIdx.x & 63;
  int idx = threadIdx.x;
  if (idx < 4 * DD) {
    int ee = blockIdx.x * 4 + idx / DD;
    xs[idx / DD][idx % DD] = cur[(size_t)src[ee] * DP + idx % DD];
  }
  __syncthreads();
  int e = blockIdx.x * 4 + grp;
  if (t < DD) {
    const __hip_bfloat162* w2 = reinterpret_cast<const __hip_bfloat162*>(
        We + (size_t)e * WEP + t * DD);  // contiguous 84B per output
    const float* x = xs[grp];
    float acc = 0.f;
#pragma unroll
    for (int i = 0; i < DD / 2; ++i) {
      float2 wf = __bfloat1622float2(w2[i]);
      acc += x[2 * i] * wf.x + x[2 * i + 1] * wf.y;
    }
    atomicAdd(&msg[(size_t)dst[e] * DP + t], acc);
  }
}

// out2 = concat([relu(msg), out]) @ msg_W^T + msg_b
__global__ void msg_combine(const float* __restrict__ msg,
                            const float* __restrict__ out,
                            const float* __restrict__ Wm,
                            const float* __restrict__ bm,
                            float* __restrict__ out2) {
  int idx = blockIdx.x * blockDim.x + threadIdx.x;
  if (idx >= NN * DD) return;
  int n = idx / DD, o = idx % DD;
  const float* w = Wm + (size_t)o * D2D;
  const float* mr = msg + (size_t)n * DP;
  const float* xr = out + (size_t)n * DP;
  float acc = bm[o];
  for (int j = 0; j < DD; ++j) acc += fmaxf(mr[j], 0.f) * w[j];
  for (int j = 0; j < DD; ++j) acc += xr[j] * w[DD + j];
  out2[(size_t)n * DP + o] = acc;
}

__global__ void add_init(const float* __restrict__ cur,
                         const float* __restrict__ nf, float* __restrict__ h) {
  int idx = blockIdx.x * blockDim.x + threadIdx.x;
  if (idx >= NN * DP) return;
  int n = idx / DP, c = idx % DP;
  h[idx] = (c < DD) ? cur[idx] + nf[(size_t)n * DD + c] : 0.f;
}

// ----------------------- imap = hu @ hv^T ; fused len_map / tanh / outputs
__global__ __launch_bounds__(256) void imap_gemm(
    const float* __restrict__ hu, const float* __restrict__ hv,
    const float* __restrict__ lu, const float* __restrict__ lv,
    float* __restrict__ ret, float* __restrict__ imt) {
  int wave = threadIdx.x >> 5, lane = threadIdx.x & 31;
  int half = lane >> 4, l15 = lane & 15;
  int i0 = blockIdx.x * 16;
  int j0 = (blockIdx.y * 8 + wave) * 16;
  v8f acc = {};
  const size_t arow = (size_t)(i0 + l15) * DP;
  const size_t brow = (size_t)(j0 + l15) * DP;
#pragma unroll
  for (int k0 = 0; k0 < DP; k0 += 4) {
    int kb = k0 + half * 2;
    v2f a, b;
    a[0] = hu[arow + kb];
    a[1] = hu[arow + kb + 1];
    b[0] = hv[brow + kb];
    b[1] = hv[brow + kb + 1];
    acc = wmma4(a, b, acc);
  }
  int j = j0 + l15;
  float lvj = lv[j];
#pragma unroll
  for (int r = 0; r < 8; ++r) {
    int i = i0 + r + half * 8;
    float lm = lu[i] * lvj;
    float d = acc[r];
    ret[(size_t)i * NN + j] = lm * d;
    imt[(size_t)i * NN + j] = lm * tanhf(d);
  }
}

// ------------------- solute_prime = imt @ hv  -> fu[:, 42..83]  (K = 8192)
__global__ __launch_bounds__(256) void sp_gemm(const float* __restrict__ imt,
                                               const float* __restrict__ hv,
                                               float* __restrict__ fu) {
  int wave = threadIdx.x >> 5, lane = threadIdx.x & 31;
  int half = lane >> 4, l15 = lane & 15;
  int i0 = (blockIdx.x * 8 + wave) * 16;
  v8f acc0 = {}, acc1 = {}, acc2 = {};
  const size_t arow = (size_t)(i0 + l15) * NN;
  for (int k0 = 0; k0 < NN; k0 += 4) {
    int kb = k0 + half * 2;
    if ((k0 & 15) == 0)  // keep the 268MB A-stream ~2KB ahead of the WMMAs
      __builtin_prefetch(&imt[arow + kb + 512], 0, 3);
    v2f a;
    a[0] = imt[arow + kb];
    a[1] = imt[arow + kb + 1];
    const float* h0 = hv + (size_t)kb * DP;
    const float* h1p = hv + (size_t)(kb + 1) * DP;
    v2f b;
    b[0] = h0[l15];      b[1] = h1p[l15];      acc0 = wmma4(a, b, acc0);
    b[0] = h0[16 + l15]; b[1] = h1p[16 + l15]; acc1 = wmma4(a, b, acc1);
    b[0] = h0[32 + l15]; b[1] = h1p[32 + l15]; acc2 = wmma4(a, b, acc2);
  }
  for (int tile = 0; tile < 3; ++tile) {
    int o = tile * 16 + l15;
    if (o >= DD) continue;
    v8f& a = tile == 0 ? acc0 : (tile == 1 ? acc1 : acc2);
#pragma unroll
    for (int r = 0; r < 8; ++r) {
      int i = i0 + r + half * 8;
      fu[(size_t)i * D2D + DD + o] = a[r];
    }
  }
}

// ----- solvent_prime = imt^T @ hu computed as C' = hu^T @ imt (coalesced B)
__global__ __launch_bounds__(256) void svp_gemm(const float* __restrict__ imt,
                                                const float* __restrict__ hu,
                                                float* __restrict__ fv) {
  int wave = threadIdx.x >> 5, lane = threadIdx.x & 31;
  int half = lane >> 4, l15 = lane & 15;
  int j0 = (blockIdx.x * 8 + wave) * 16;
  v8f acc0 = {}, acc1 = {}, acc2 = {};
  for (int k0 = 0; k0 < NN; k0 += 4) {
    int kb = k0 + half * 2;
    if ((k0 & 15) == 0 && k0 + 128 < NN)  // prefetch B-stream rows ahead
      __builtin_prefetch(&imt[(size_t)(kb + 128) * NN + j0 + l15], 0, 3);
    v2f b;
    b[0] = imt[(size_t)kb * NN + j0 + l15];
    b[1] = imt[(size_t)(kb + 1) * NN + j0 + l15];
    const float* g0 = hu + (size_t)kb * DP;
    const float* g1 = hu + (size_t)(kb + 1) * DP;
    v2f a;
    a[0] = g0[l15];      a[1] = g1[l15];      acc0 = wmma4(a, b, acc0);
    a[0] = g0[16 + l15]; a[1] = g1[16 + l15]; acc1 = wmma4(a, b, acc1);
    a[0] = g0[32 + l15]; a[1] = g1[32 + l15]; acc2 = wmma4(a, b, acc2);
  }
  int j = j0 + l15;
  for (int tile = 0; tile < 3; ++tile) {
    v8f& a = tile == 0 ? acc0 : (tile == 1 ? acc1 : acc2);
#pragma unroll
    for (int r = 0; r < 8; ++r) {
      int o = tile * 16 + r + half * 8;
      if (o < DD) fv[(size_t)j * D2D + DD + o] = a[r];
    }
  }
}

__global__ void copy_half(const float* __restrict__ h /*N x DP*/,
                          float* __restrict__ f /*N x D2D*/) {
  int idx = blockIdx.x * blockDim.x + threadIdx.x;
  if (idx >= NN * DD) return;
  int n = idx / DD, c = idx % DD;
  f[(size_t)n * D2D + c] = h[(size_t)n * DP + c];
}

// ------------------------------------------------- Set2Set (one WG / graph)
__global__ __launch_bounds__(256) void set2set(
    const float* __restrict__ feat /*N x 84*/, const float* __restrict__ Wih,
    const float* __restrict__ Whh, const float* __restrict__ bih,
    const float* __restrict__ bhh, float* __restrict__ pout /*B x 168*/) {
  int g = blockIdx.x, tid = threadIdx.x;
  __shared__ float q[168], h[D2D], c[D2D], gates[336];
  __shared__ float ev[NPGD], red[NPGD], r[D2D];
  if (tid < 168) q[tid] = 0.f;
  if (tid < D2D) { h[tid] = 0.f; c[tid] = 0.f; }
  __syncthreads();
  for (int it = 0; it < 2; ++it) {
    for (int o = tid; o < 336; o += 256) {
      float acc = bih[o] + bhh[o];
      const float* wi = Wih + (size_t)o * 168;
      for (int k = 0; k < 168; ++k) acc += q[k] * wi[k];
      const float* wh = Whh + (size_t)o * D2D;
      for (int k = 0; k < D2D; ++k) acc += h[k] * wh[k];
      gates[o] = acc;
    }
    __syncthreads();
    if (tid < D2D) {
      float ig = 1.f / (1.f + __expf(-gates[tid]));
      float fg = 1.f / (1.f + __expf(-gates[D2D + tid]));
      float gg = tanhf(gates[2 * D2D + tid]);
      float og = 1.f / (1.f + __expf(-gates[3 * D2D + tid]));
      float cn = fg * c[tid] + ig * gg;
      c[tid] = cn;
      h[tid] = og * tanhf(cn);
    }
    __syncthreads();
    // attention over the 256 nodes of this graph
    const float* fr = feat + ((size_t)g * NPGD + tid) * D2D;
    float e = 0.f;
    for (int k = 0; k < D2D; ++k) e += fr[k] * h[k];
    red[tid] = e;
    __syncthreads();
    for (int s = 128; s > 0; s >>= 1) {
      if (tid < s) red[tid] = fmaxf(red[tid], red[tid + s]);
      __syncthreads();
    }
    float emax = red[0];
    __syncthreads();
    float ex = __expf(e - emax);
    ev[tid] = ex;
    red[tid] = ex;
    __syncthreads();
    for (int s = 128; s > 0; s >>= 1) {
      if (tid < s) red[tid] += red[tid + s];
      __syncthreads();
    }
    float esum = red[0];
    __syncthreads();
    if (tid < D2D) {
      float acc = 0.f;
      for (int n = 0; n < NPGD; ++n)
        acc += ev[n] * feat[((size_t)g * NPGD + n) * D2D + tid];
      r[tid] = acc / esum;
    }
    __syncthreads();
    if (tid < D2D) { q[tid] = h[tid]; q[D2D + tid] = r[tid]; }
    __syncthreads();
  }
  if (tid < 168) pout[(size_t)g * 168 + tid] = q[tid];
}

// -------------------------------------------------- final MLP (one WG / g)
__global__ __launch_bounds__(256) void mlp(
    const float* __restrict__ pu, const float* __restrict__ pv,
    const float* __restrict__ W1, const float* __restrict__ b1,
    const float* __restrict__ W2, const float* __restrict__ b2,
    const float* __restrict__ W3, const float* __restrict__ b3,
    float* __restrict__ pred) {
  int g = blockIdx.x, tid = threadIdx.x;
  __shared__ float x[336], a1[256], a2[128], red[128];
  if (tid < 168) {
    x[tid] = pu[(size_t)g * 168 + tid];
    x[168 + tid] = pv[(size_t)g * 168 + tid];
  }
  __syncthreads();
  {
    float acc = b1[tid];
    const float* w = W1 + (size_t)tid * 336;
    for (int k = 0; k < 336; ++k) acc += x[k] * w[k];
    a1[tid] = fmaxf(acc, 0.f);
  }
  __syncthreads();
  if (tid < 128) {
    float acc = b2[tid];
    const float* w = W2 + (size_t)tid * 256;
    for (int k = 0; k < 256; ++k) acc += a1[k] * w[k];
    a2[tid] = fmaxf(acc, 0.f);
  }
  __syncthreads();
  if (tid < 128) red[tid] = a2[tid] * W3[tid];
  __syncthreads();
  for (int s = 64; s > 0; s >>= 1) {
    if (tid < s) red[tid] += red[tid + s];
    __syncthreads();
  }
  if (tid == 0) pred[g] = red[0] + b3[0];
}

// =========================================================== host dispatch
struct GP {
  const float *lin0_W, *lin0_b, *en1_W, *en1_b, *en2_W, *en2_b, *res_W,
      *conv_b, *msg_W, *msg_b;
};

static inline unsigned cdiv(size_t a, unsigned b) {
  return (unsigned)((a + b - 1) / b);
}

static void run_gather(const GP& P, const float* nf, const float* ef,
                       const int* src, const int* dst, float* h1, float* W2p,
                       __hip_bfloat16* We, float* bufA, float* bufB, float* msg,
                       float* hout, hipStream_t s) {
  fillz<<<2048, 256, 0, s>>>(h1, (size_t)EE * DP);
  dense<<<cdiv((size_t)EE * DD, 256), 256, 0, s>>>(ef, EIND, P.en1_W, P.en1_b,
                                                   h1, DP, EE, DD, EIND, 1);
  pad_w2<<<cdiv((size_t)WEP * KP2, 256), 256, 0, s>>>(P.en2_W, W2p);
  we_gemm<<<dim3(EE / 16, WEP / 16 / 8), 256, 0, s>>>(h1, W2p, P.en2_b, We);
  fillz<<<256, 256, 0, s>>>(bufA, (size_t)NN * DP);
  fillz<<<256, 256, 0, s>>>(bufB, (size_t)NN * DP);
  dense<<<cdiv((size_t)NN * DD, 256), 256, 0, s>>>(nf, DD, P.lin0_W, P.lin0_b,
                                                   bufA, DP, NN, DD, DD, 1);
  float* cur = bufA;
  float* nxt = bufB;
  for (int step = 0; step < 6; ++step) {
    // msg <- residual + conv bias, then edges atomically accumulate on top
    dense<<<cdiv((size_t)NN * DD, 256), 256, 0, s>>>(cur, DP, P.res_W, P.conv_b,
                                                     msg, DP, NN, DD, DD, 0);
    edge_msg<<<EE / 4, 256, 0, s>>>(cur, We, src, dst, msg);
    msg_combine<<<cdiv((size_t)NN * DD, 256), 256, 0, s>>>(msg, cur, P.msg_W,
                                                           P.msg_b, nxt);
    float* t = cur; cur = nxt; nxt = t;
  }
  add_init<<<cdiv((size_t)NN * DP, 256), 256, 0, s>>>(cur, nf, hout);
}

extern "C" void kernel_launch(void* const* d_in, const int* in_sizes, int n_in,
                              void* d_out, int out_size, void* d_ws,
                              size_t ws_size, hipStream_t stream) {
  (void)in_sizes; (void)n_in; (void)out_size; (void)ws_size;
  const float* n_feat_u = (const float*)d_in[0];
  const float* e_feat_u = (const float*)d_in[1];
  const float* n_feat_v = (const float*)d_in[2];
  const float* e_feat_v = (const float*)d_in[3];
  const int* src_u = (const int*)d_in[4];
  const int* dst_u = (const int*)d_in[5];
  const int* src_v = (const int*)d_in[6];
  const int* dst_v = (const int*)d_in[7];
  // seg_u (8) / seg_v (9): graphs are contiguous blocks of 256 nodes
  const float* solute_len = (const float*)d_in[10];
  const float* solvent_len = (const float*)d_in[11];
  auto gp = [&](int b) {
    GP g = {(const float*)d_in[b + 0], (const float*)d_in[b + 1],
            (const float*)d_in[b + 2], (const float*)d_in[b + 3],
            (const float*)d_in[b + 4], (const float*)d_in[b + 5],
            (const float*)d_in[b + 6], (const float*)d_in[b + 7],
            (const float*)d_in[b + 8], (const float*)d_in[b + 9]};
    return g;
  };
  GP sol = gp(12), slv = gp(22);
  const float* Wih_u = (const float*)d_in[32];
  const float* Whh_u = (const float*)d_in[33];
  const float* bih_u = (const float*)d_in[34];
  const float* bhh_u = (const float*)d_in[35];
  const float* Wih_v = (const float*)d_in[36];
  const float* Whh_v = (const float*)d_in[37];
  const float* bih_v = (const float*)d_in[38];
  const float* bhh_v = (const float*)d_in[39];
  const float* fc1_W = (const float*)d_in[40];
  const float* fc1_b = (const float*)d_in[41];
  const float* fc2_W = (const float*)d_in[42];
  const float* fc2_b = (const float*)d_in[43];
  const float* fc3_W = (const float*)d_in[44];
  const float* fc3_b = (const float*)d_in[45];

  // -------- workspace carve-up
  char* w = (char*)d_ws;
  size_t off = 0;
  auto alloc = [&](size_t bytes) {
    void* p = w + off;
    off += (bytes + 255) & ~(size_t)255;
    return p;
  };
  float* h1 = (float*)alloc((size_t)EE * DP * 4);
  float* W2p = (float*)alloc((size_t)WEP * KP2 * 4);
  __hip_bfloat16* We = (__hip_bfloat16*)alloc((size_t)EE * WEP * 2);
  float* bufA = (float*)alloc((size_t)NN * DP * 4);
  float* bufB = (float*)alloc((size_t)NN * DP * 4);
  float* msg = (float*)alloc((size_t)NN * DP * 4);
  float* hu = (float*)alloc((size_t)NN * DP * 4);
  float* hv = (float*)alloc((size_t)NN * DP * 4);
  float* fu = (float*)alloc((size_t)NN * D2D * 4);
  float* fv = (float*)alloc((size_t)NN * D2D * 4);
  float* pu = (float*)alloc((size_t)BB * 168 * 4);
  float* pv = (float*)alloc((size_t)BB * 168 * 4);
  float* imt = (float*)alloc((size_t)NN * NN * 4);

  float* pred = (float*)d_out;          // [32]
  float* ret = (float*)d_out + BB;      // ret_imap [8192 x 8192]

  // -------- GatherModel for solute (u) then solvent (v); We buffer reused
  run_gather(sol, n_feat_u, e_feat_u, src_u, dst_u, h1, W2p, We, bufA, bufB,
             msg, hu, stream);
  run_gather(slv, n_feat_v, e_feat_v, src_v, dst_v, h1, W2p, We, bufA, bufB,
             msg, hv, stream);

  // -------- interaction map (fused len_map * {imap, tanh(imap)})
  imap_gemm<<<dim3(NN / 16, NN / 16 / 8), 256, 0, stream>>>(
      hu, hv, solute_len, solvent_len, ret, imt);

  // -------- fu = [hu, imt @ hv], fv = [hv, imt^T @ hu]
  copy_half<<<cdiv((size_t)NN * DD, 256), 256, 0, stream>>>(hu, fu);
  copy_half<<<cdiv((size_t)NN * DD, 256), 256, 0, stream>>>(hv, fv);
  sp_gemm<<<NN / 16 / 8, 256, 0, stream>>>(imt, hv, fu);
  svp_gemm<<<NN / 16 / 8, 256, 0, stream>>>(imt, hu, fv);

  // -------- set2set readouts + final MLP
  set2set<<<BB, 256, 0, stream>>>(fu, Wih_u, Whh_u, bih_u, bhh_u, pu);
  set2set<<<BB, 256, 0, stream>>>(fv, Wih_v, Whh_v, bih_v, bhh_v, pv);
  mlp<<<BB, 256, 0, stream>>>(pu, pv, fc1_W, fc1_b, fc2_W, fc2_b, fc3_W, fc3_b,
                              pred);
}